// Inv_Loss_Embed_58059367907729
// MI455X (gfx1250) — compile-verified
//
#include <hip/hip_runtime.h>
#include <hip/hip_bf16.h>

typedef __attribute__((ext_vector_type(2))) float v2f;
typedef __attribute__((ext_vector_type(8))) float v8f;

#define NS      8192
#define EMBED   64
#define INV_TAU 5.0f

// ---------------------------------------------------------------------------
// Kernel 1: gather selected rows into contiguous [2][8192][64] buffer and
// compute diag[row] = ||e||^2 / tau.  16 lanes per row, float4 each.
// ---------------------------------------------------------------------------
__global__ __launch_bounds__(256) void gather_diag_kernel(
    const float* __restrict__ all_users, const float* __restrict__ all_items,
    const int* __restrict__ uidx, const int* __restrict__ iidx,
    float* __restrict__ gathered, float* __restrict__ diag) {
  int t = blockIdx.x * blockDim.x + threadIdx.x;
  int slot  = t >> 4;          // 0 .. 16383  (global row slot)
  int lane4 = t & 15;          // 16 threads per row
  if (slot >= 2 * NS) return;
  int m = slot >> 13;          // 0 = users, 1 = items
  int r = slot & (NS - 1);
  const float* src = (m == 0) ? all_users : all_items;
  long long srow   = (m == 0) ? (long long)uidx[r] : (long long)iidx[r];

  const float4 v = ((const float4*)(src + srow * EMBED))[lane4];
  ((float4*)(gathered + (size_t)slot * EMBED))[lane4] = v;

  float sq = v.x * v.x + v.y * v.y + v.z * v.z + v.w * v.w;
  #pragma unroll
  for (int off = 1; off < 16; off <<= 1)
    sq += __shfl_xor(sq, off, 32);
  if (lane4 == 0) diag[slot] = sq * INV_TAU;
}

// ---------------------------------------------------------------------------
// Kernel 2: fused Gram-matrix + streaming logsumexp.
// One wave owns a 16-row tile; streams 512 column tiles, 2 per iteration
// (two independent 16-deep V_WMMA_F32_16X16X4_F32 accumulation chains).
// Per-lane online softmax state: 8 slots (one per C/D VGPR = one row M),
// single-exp update:  e = exp(-|s - m|);  d>0 ? sum*e+1 : sum+e.
// ---------------------------------------------------------------------------
__global__ __launch_bounds__(128) void ssm_lse_kernel(
    const float* __restrict__ gathered, const float* __restrict__ diag,
    float* __restrict__ perrow) {
  const int lane    = threadIdx.x & 31;
  const int wave    = threadIdx.x >> 5;
  const int mat     = blockIdx.x >> 7;                  // 0 users, 1 items
  const int rowTile = ((blockIdx.x & 127) << 2) + wave; // 0..511

  const float* E = gathered + (size_t)mat * NS * EMBED;
  const int rbase  = rowTile * 16;
  const int mrow   = lane & 15;          // M (and N for B-frag)
  const int colOff = (lane >> 4) * 2;    // K sub-offset within 4-chunk

  // Preload A fragments for this 16x64 row tile: a[k] covers K = 4k .. 4k+3.
  v2f a[16];
  #pragma unroll
  for (int k = 0; k < 16; ++k)
    a[k] = *(const v2f*)(E + (size_t)(rbase + mrow) * EMBED + 4 * k + colOff);

  float mx[8], sm[8];
  #pragma unroll
  for (int g = 0; g < 8; ++g) { mx[g] = -__builtin_inff(); sm[g] = 0.0f; }

  for (int ct = 0; ct < NS / 16; ct += 2) {
    const float* B0 = E + (size_t)(ct * 16 + mrow) * EMBED + colOff;
    const float* B1 = B0 + 16 * EMBED;

    v2f b0[16], b1[16];
    #pragma unroll
    for (int k = 0; k < 16; ++k) b0[k] = *(const v2f*)(B0 + 4 * k);
    #pragma unroll
    for (int k = 0; k < 16; ++k) b1[k] = *(const v2f*)(B1 + 4 * k);

    v8f c0 = {}, c1 = {};
    #pragma unroll
    for (int k = 0; k < 16; ++k) {
      c0 = __builtin_amdgcn_wmma_f32_16x16x4_f32(
               false, a[k], false, b0[k], (short)0, c0, false, false);
      c1 = __builtin_amdgcn_wmma_f32_16x16x4_f32(
               false, a[k], false, b1[k], (short)0, c1, false, false);
    }

    // Lane-local online softmax update, one exp per score.
    #pragma unroll
    for (int g = 0; g < 8; ++g) {
      float s = c0[g] * INV_TAU;
      float d = s - mx[g];
      float e = __expf(-__builtin_fabsf(d));
      sm[g] = (d > 0.0f) ? (sm[g] * e + 1.0f) : (sm[g] + e);
      mx[g] = fmaxf(mx[g], s);
    }
    #pragma unroll
    for (int g = 0; g < 8; ++g) {
      float s = c1[g] * INV_TAU;
      float d = s - mx[g];
      float e = __expf(-__builtin_fabsf(d));
      sm[g] = (d > 0.0f) ? (sm[g] * e + 1.0f) : (sm[g] + e);
      mx[g] = fmaxf(mx[g], s);
    }
  }

  // Combine (max,sum) across the 16 lanes sharing each row, then write
  // lse - diag.  Rows g (lanes 0-15) and g+8 (lanes 16-31) per slot g.
  #pragma unroll
  for (int g = 0; g < 8; ++g) {
    float m = mx[g], s = sm[g];
    #pragma unroll
    for (int off = 1; off < 16; off <<= 1) {
      float m2 = __shfl_xor(m, off, 32);
      float s2 = __shfl_xor(s, off, 32);
      float mn = fmaxf(m, m2);
      s = s * __expf(m - mn) + s2 * __expf(m2 - mn);
      m = mn;
    }
    if ((lane & 15) == 0) {
      int r = rbase + g + ((lane >> 4) << 3);
      int idx = mat * NS + r;
      perrow[idx] = m + __logf(s) - diag[idx];
    }
  }
}

// ---------------------------------------------------------------------------
// Kernel 3: deterministic final reduction -> [inv_loss, loss_u, loss_i].
// ---------------------------------------------------------------------------
__global__ __launch_bounds__(256) void reduce_kernel(
    const float* __restrict__ perrow, float* __restrict__ out) {
  __shared__ float su[256];
  __shared__ float si[256];
  int t = threadIdx.x;
  float au = 0.0f, ai = 0.0f;
  for (int i = t; i < NS; i += 256) {
    au += perrow[i];
    ai += perrow[NS + i];
  }
  su[t] = au; si[t] = ai;
  __syncthreads();
  for (int s = 128; s > 0; s >>= 1) {
    if (t < s) { su[t] += su[t + s]; si[t] += si[t + s]; }
    __syncthreads();
  }
  if (t == 0) {
    float lu = su[0] * (1.0f / NS);
    float li = si[0] * (1.0f / NS);
    out[0] = lu + li;
    out[1] = lu;
    out[2] = li;
  }
}

// ---------------------------------------------------------------------------
extern "C" void kernel_launch(void* const* d_in, const int* in_sizes, int n_in,
                              void* d_out, int out_size, void* d_ws, size_t ws_size,
                              hipStream_t stream) {
  const float* all_users = (const float*)d_in[0];
  const float* all_items = (const float*)d_in[1];
  const int*   uidx      = (const int*)d_in[2];
  const int*   iidx      = (const int*)d_in[3];
  float* out = (float*)d_out;

  float* gathered = (float*)d_ws;                         // 2*8192*64 floats
  float* diag     = gathered + (size_t)2 * NS * EMBED;    // 2*8192 floats
  float* perrow   = diag + 2 * NS;                        // 2*8192 floats

  // 1) gather + diag: 16384 rows * 16 lanes = 262144 threads
  gather_diag_kernel<<<(2 * NS * 16) / 256, 256, 0, stream>>>(
      all_users, all_items, uidx, iidx, gathered, diag);

  // 2) fused WMMA + streaming logsumexp: 2 matrices * 128 blocks * 4 waves
  ssm_lse_kernel<<<256, 128, 0, stream>>>(gathered, diag, perrow);

  // 3) final deterministic reduction
  reduce_kernel<<<1, 256, 0, stream>>>(perrow, out);
}